// LSTMStocksModule_63410897158719
// MI455X (gfx1250) — compile-verified
//
#include <hip/hip_runtime.h>

// LSTM: S=512, B=2048, D=64, H=2 (4H = 8 gates)
// Phase 1: x_proj[s,b,g] = x[s,b,:]·W_ih[g,:] + b_ih[g] + b_hh[g]  via V_WMMA_F32_16X16X4_F32
// Phase 2: per-batch sequential recurrence, 2048 independent chains.

typedef __attribute__((ext_vector_type(2))) float v2f;
typedef __attribute__((ext_vector_type(8))) float v8f;

#define SEQ   512
#define BATCH 2048
#define DIN   64
#define NG    8                     // 4*H gate columns (padded to 16 in WMMA N)
#define ROWS  (SEQ * BATCH)         // 1,048,576 flat (s,b) rows
#define TILES (ROWS / 16)           // 65,536 M-tiles of 16 rows
#define GEMM_BLOCKS     512
#define WAVES_PER_BLOCK 4
#define TILES_PER_WAVE  (TILES / (GEMM_BLOCKS * WAVES_PER_BLOCK))   // 32

__global__ __launch_bounds__(128) void lstm_xproj_wmma(
    const float* __restrict__ x,    // [S,B,D] contiguous -> flat [ROWS, 64]
    const float* __restrict__ Wih,  // [8, 64]
    const float* __restrict__ bih,  // [8]
    const float* __restrict__ bhh,  // [8]
    float* __restrict__ xp)         // [ROWS, 8]
{
  __shared__ float lds[WAVES_PER_BLOCK * 16 * DIN];   // 16 KB: 4 KB tile per wave
  const int lane = threadIdx.x & 31;
  const int wid  = threadIdx.x >> 5;
  const int n    = lane & 15;       // N index (gate column) for B/C/D operands
  const int hi   = lane >> 4;       // half-wave selector (K sub-pair / M+8)

  // ---- Preload B operand (W_ih^T, K=64 as 16 chunks of K=4, N padded to 16) ----
  // 32-bit B 4x16 layout mirrors A: lanes 0-15 = N, VGPR0/1 hold K={0,1} (lo half)
  // and K={2,3} (hi half).
  v2f breg[16];
#pragma unroll
  for (int k = 0; k < 16; ++k) {
    float bx = 0.f, by = 0.f;
    if (n < NG) {
      bx = Wih[n * DIN + 4 * k + 2 * hi];
      by = Wih[n * DIN + 4 * k + 2 * hi + 1];
    }
    breg[k].x = bx;
    breg[k].y = by;
  }
  // Fused bias: C initialized to (b_ih+b_hh)[N], broadcast along M.
  const float bias = (n < NG) ? (bih[n] + bhh[n]) : 0.f;

  float* myLds = lds + wid * (16 * DIN);
  const int gw = blockIdx.x * WAVES_PER_BLOCK + wid;

  for (int t = 0; t < TILES_PER_WAVE; ++t) {
    const int  tile = gw * TILES_PER_WAVE + t;
    const long r0   = (long)tile * 16;

    // Stage 16x64 fp32 tile into LDS: 256 float4, coalesced b128 loads.
    const float4* src = (const float4*)(x + r0 * DIN);
#pragma unroll
    for (int j = 0; j < 8; ++j) {
      ((float4*)myLds)[lane + 32 * j] = src[lane + 32 * j];
    }
    __syncthreads();

    v8f acc;
#pragma unroll
    for (int v = 0; v < 8; ++v) acc[v] = bias;

    // A operand per K-chunk from LDS per ISA layout:
    // lane m=lane&15 holds row M=m; lo half K={4k,4k+1}, hi half K={4k+2,4k+3}.
#pragma unroll
    for (int k = 0; k < 16; ++k) {
      v2f a = *(const v2f*)(myLds + n * DIN + 4 * k + 2 * hi);
      acc = __builtin_amdgcn_wmma_f32_16x16x4_f32(
          /*neg_a=*/false, a, /*neg_b=*/false, breg[k],
          /*c_mod=*/(short)0, acc, /*reuse_a=*/false, /*reuse_b=*/false);
    }
    __syncthreads();   // LDS reused next iteration

    // C/D layout: VGPR v, lane l -> M = v + 8*hi, N = lane&15. Store 8 real cols.
    if (n < NG) {
#pragma unroll
      for (int v = 0; v < 8; ++v) {
        const long m = v + 8 * hi;
        xp[(r0 + m) * NG + n] = acc[v];
      }
    }
  }
}

__device__ __forceinline__ float sigf(float v) { return 1.0f / (1.0f + __expf(-v)); }
__device__ __forceinline__ float tanhfast(float v) { return 2.0f * sigf(2.0f * v) - 1.0f; }

__global__ __launch_bounds__(256) void lstm_recurrence(
    const float* __restrict__ xp,   // [S, B, 8]
    const float* __restrict__ Whh,  // [8, 2]
    float* __restrict__ out)        // h_n [B,2] then c_n [B,2]
{
  const int b = blockIdx.x * blockDim.x + threadIdx.x;   // batch chain id

  float w[16];
#pragma unroll
  for (int i = 0; i < 16; ++i) w[i] = Whh[i];

  float h0 = 0.f, h1 = 0.f, c0 = 0.f, c1 = 0.f;

  const float4* p = (const float4*)(xp + (long)b * NG);
  const int stride4 = BATCH * NG / 4;   // float4 stride between timesteps

  // Manual prefetch pipeline: next step's gates load independent of recurrence.
  float4 ga = p[0], gb = p[1];
  for (int s = 0; s < SEQ; ++s) {
    float4 na = {0.f, 0.f, 0.f, 0.f}, nb = {0.f, 0.f, 0.f, 0.f};
    if (s + 1 < SEQ) {
      na = p[(long)(s + 1) * stride4];
      nb = p[(long)(s + 1) * stride4 + 1];
    }
    // torch gate order: i, f, g, o (each H=2 wide)
    const float gi0 = ga.x + h0 * w[0]  + h1 * w[1];
    const float gi1 = ga.y + h0 * w[2]  + h1 * w[3];
    const float gf0 = ga.z + h0 * w[4]  + h1 * w[5];
    const float gf1 = ga.w + h0 * w[6]  + h1 * w[7];
    const float gg0 = gb.x + h0 * w[8]  + h1 * w[9];
    const float gg1 = gb.y + h0 * w[10] + h1 * w[11];
    const float go0 = gb.z + h0 * w[12] + h1 * w[13];
    const float go1 = gb.w + h0 * w[14] + h1 * w[15];

    c0 = sigf(gf0) * c0 + sigf(gi0) * tanhfast(gg0);
    c1 = sigf(gf1) * c1 + sigf(gi1) * tanhfast(gg1);
    h0 = sigf(go0) * tanhfast(c0);
    h1 = sigf(go1) * tanhfast(c1);

    ga = na; gb = nb;
  }

  out[b * 2 + 0] = h0;
  out[b * 2 + 1] = h1;
  out[2 * BATCH + b * 2 + 0] = c0;
  out[2 * BATCH + b * 2 + 1] = c1;
}

extern "C" void kernel_launch(void* const* d_in, const int* in_sizes, int n_in,
                              void* d_out, int out_size, void* d_ws, size_t ws_size,
                              hipStream_t stream) {
  (void)in_sizes; (void)n_in; (void)out_size; (void)ws_size;
  const float* x   = (const float*)d_in[0];
  const float* Wih = (const float*)d_in[1];
  const float* Whh = (const float*)d_in[2];
  const float* bih = (const float*)d_in[3];
  const float* bhh = (const float*)d_in[4];

  float* xp  = (float*)d_ws;   // needs 512*2048*8*4 = 32 MB scratch
  float* out = (float*)d_out;

  lstm_xproj_wmma<<<GEMM_BLOCKS, 128, 0, stream>>>(x, Wih, bih, bhh, xp);
  lstm_recurrence<<<BATCH / 256, 256, 0, stream>>>(xp, Whh, out);
}